// SelfAttention_85676007621083
// MI455X (gfx1250) — compile-verified
//
#include <hip/hip_runtime.h>

typedef __attribute__((ext_vector_type(2))) float v2f;
typedef __attribute__((ext_vector_type(4))) float v4f;
typedef __attribute__((ext_vector_type(8))) float v8f;
typedef __attribute__((ext_vector_type(4))) int   i4;

#define NUM_HEADS 12
#define HEAD_DIM  64
#define SEQ       2048
#define HID       768
#define NBITS     5
#define NEGBIG    (-1.0e4f)
#define BATCH     2

// D = A(16x4,f32) * B(4x16,f32) + C(16x16,f32), full fp32 on the matrix pipe.
static __device__ __forceinline__ v8f wmma_f32(v2f a, v2f b, v8f c) {
  return __builtin_amdgcn_wmma_f32_16x16x4_f32(false, a, false, b, (short)0, c,
                                               false, false);
}

// --- gfx1250 async global->LDS (ASYNCcnt) helpers ---------------------------
static __device__ __forceinline__ void async_copy16(const float* g, float* l) {
#if __has_builtin(__builtin_amdgcn_global_load_async_to_lds_b128)
  __builtin_amdgcn_global_load_async_to_lds_b128(
      (__attribute__((address_space(1))) i4*)(void*)g,
      (__attribute__((address_space(3))) i4*)(void*)l, 0, 0);
#else
  unsigned long long ga = (unsigned long long)g;
  unsigned lo = (unsigned)(size_t)l;  // flat LDS addr: low 32 bits = LDS offset
  asm volatile("global_load_async_to_lds_b128 %0, %1, off"
               :: "v"(lo), "v"(ga) : "memory");
#endif
}

template <int N>
static __device__ __forceinline__ void wait_async() {
#if __has_builtin(__builtin_amdgcn_s_wait_asynccnt)
  __builtin_amdgcn_s_wait_asynccnt(N);
#else
  asm volatile("s_wait_asynccnt %0" :: "n"(N) : "memory");
#endif
}

// ---------------------------------------------------------------------------
// Kernel 1: Q/V projection GEMM.  One wave per 16x64 output tile.
//   out[b,h,s,d] = (hs @ W + bias) split into heads, head-major layout.
// grid = (4096/16, 768/64, 2{Q,V}), block = 32.
// ---------------------------------------------------------------------------
__global__ __launch_bounds__(32) void qv_proj_kernel(
    const float* __restrict__ hs,
    const float* __restrict__ Wq, const float* __restrict__ bq,
    const float* __restrict__ Wv, const float* __restrict__ bv,
    float* __restrict__ Qh, float* __restrict__ Vh) {
  const int lane = threadIdx.x & 31;
  const int half = lane >> 4;      // 0: K=0,1 / rows 0-7 of C; 1: K=2,3 / rows 8-15
  const int l16  = lane & 15;

  const int row0 = blockIdx.x * 16;   // row in [B*S]
  const int head = blockIdx.y;        // 64-wide column group == one head
  const int col0 = head * 64;
  const bool isV = (blockIdx.z != 0);

  const float* __restrict__ W    = isV ? Wv : Wq;
  const float* __restrict__ bias = isV ? bv : bq;
  float* __restrict__ out        = isV ? Vh : Qh;

  const float* arow = hs + (size_t)(row0 + l16) * HID;

  v8f acc[4];
  acc[0] = v8f{}; acc[1] = v8f{}; acc[2] = v8f{}; acc[3] = v8f{};

  for (int k = 0; k < HID; k += 4) {
    const int kk = k + half * 2;
    const v2f a = *(const v2f*)(arow + kk);        // A[m][kk], A[m][kk+1]
    const float* wr0 = W + (size_t)kk * HID + col0;
    const float* wr1 = wr0 + HID;
#pragma unroll
    for (int t = 0; t < 4; ++t) {
      v2f b;
      b.x = wr0[t * 16 + l16];                     // B[kk  ][n]
      b.y = wr1[t * 16 + l16];                     // B[kk+1][n]
      acc[t] = wmma_f32(a, b, acc[t]);
    }
  }

#pragma unroll
  for (int t = 0; t < 4; ++t) {
    const int c = t * 16 + l16;                    // 0..63 within head
    const float bv_ = bias[col0 + c];
#pragma unroll
    for (int r = 0; r < 8; ++r) {
      const int g = row0 + half * 8 + r;           // global row in [B*S]
      const int b_ = g >> 11;                      // /SEQ
      const int s  = g & (SEQ - 1);
      out[(((size_t)b_ * NUM_HEADS + head) * SEQ + s) * HEAD_DIM + c] =
          acc[t][r] + bv_;
    }
  }
}

// ---------------------------------------------------------------------------
// Kernel 2: SimpleLSH bucket codes.  One 256-thread block per (b,h).
// ---------------------------------------------------------------------------
__global__ __launch_bounds__(256) void lsh_codes_kernel(
    const float* __restrict__ Qh, const float* __restrict__ lsh,
    int* __restrict__ codes) {
  __shared__ float sq[SEQ];
  __shared__ float red[256];
  __shared__ float s_lsh[(HEAD_DIM + 1) * NBITS];

  const int bh  = blockIdx.x;  // 0..B*NH-1
  const int tid = threadIdx.x;
  const float* qb = Qh + (size_t)bh * SEQ * HEAD_DIM;

  for (int i = tid; i < (HEAD_DIM + 1) * NBITS; i += 256) s_lsh[i] = lsh[i];

  float lmax = 0.0f;
  for (int s = tid; s < SEQ; s += 256) {
    const float* qr = qb + (size_t)s * HEAD_DIM;
    float acc = 0.0f;
#pragma unroll
    for (int d = 0; d < HEAD_DIM; d += 4) {
      v4f q4 = *(const v4f*)(qr + d);
      acc += q4.x * q4.x + q4.y * q4.y + q4.z * q4.z + q4.w * q4.w;
    }
    sq[s] = acc;
    lmax = fmaxf(lmax, acc);
  }
  red[tid] = lmax;
  __syncthreads();
  for (int off = 128; off > 0; off >>= 1) {
    if (tid < off) red[tid] = fmaxf(red[tid], red[tid + off]);
    __syncthreads();
  }
  const float M2    = red[0];
  const float inv_m = rsqrtf(M2);

  for (int s = tid; s < SEQ; s += 256) {
    const float* qr = qb + (size_t)s * HEAD_DIM;
    const float qn2 = sq[s] / M2;
    const float res = sqrtf(fmaxf(1.0f - qn2, 1e-6f));
    float acc[NBITS];
#pragma unroll
    for (int k = 0; k < NBITS; ++k) acc[k] = res * s_lsh[HEAD_DIM * NBITS + k];
    for (int d = 0; d < HEAD_DIM; ++d) {
      const float qv = qr[d] * inv_m;
#pragma unroll
      for (int k = 0; k < NBITS; ++k) acc[k] += qv * s_lsh[d * NBITS + k];
    }
    int code = 0;
#pragma unroll
    for (int k = 0; k < NBITS; ++k) code |= (acc[k] > 0.0f) ? (1 << k) : 0;
    codes[(size_t)bh * SEQ + s] = code;
  }
}

// ---------------------------------------------------------------------------
// Kernel 3: flash attention with LSH additive mask.
//   grid = (SEQ/128, NH, B), block = 256 (8 waves); wave owns 16 query rows.
//   K/V tiles double-buffered in LDS, filled by async global->LDS DMA so the
//   next tile's fetch overlaps the current tile's 32 WMMAs + softmax.
// ---------------------------------------------------------------------------
#define KSTRIDE 68   // 16x64 tile padded: bank = (n*68+c)%64 = (4n+c)%64 -> no conflict
#define PSTRIDE 18   // 16x16 P tile padded: m*18 mod 64 distinct for m=0..15

__global__ __launch_bounds__(256) void attn_kernel(
    const float* __restrict__ Qh, const float* __restrict__ Vh,
    const int* __restrict__ codes, const float* __restrict__ amask,
    float* __restrict__ out) {
  __shared__ float Kt[2][16 * KSTRIDE];
  __shared__ float Vt[2][16 * KSTRIDE];
  __shared__ float Pl[8][16 * PSTRIDE];

  const int tid  = threadIdx.x;
  const int wave = tid >> 5;
  const int lane = tid & 31;
  const int half = lane >> 4;
  const int l16  = lane & 15;

  const int b  = blockIdx.z;
  const int h  = blockIdx.y;
  const int bh = b * NUM_HEADS + h;
  const int qrow0 = blockIdx.x * 128 + wave * 16;

  const float* __restrict__ qbase = Qh + (size_t)bh * SEQ * HEAD_DIM;
  const float* __restrict__ vbase = Vh + (size_t)bh * SEQ * HEAD_DIM;
  const int*   __restrict__ cbase = codes + (size_t)bh * SEQ;
  const float* __restrict__ am    = amask + (size_t)b * SEQ;

  // Preload this wave's Q rows as A-fragments (16 k-chunks of 4).
  v2f aq[16];
  {
    const float* qr = qbase + (size_t)(qrow0 + l16) * HEAD_DIM + half * 2;
#pragma unroll
    for (int kc = 0; kc < 16; ++kc) aq[kc] = *(const v2f*)(qr + kc * 4);
  }
  // Bucket codes for the rows this lane holds in C-layout (rows half*8+r).
  int cq[8];
#pragma unroll
  for (int r = 0; r < 8; ++r) cq[r] = cbase[qrow0 + half * 8 + r];

  v8f O[4];
  O[0] = v8f{}; O[1] = v8f{}; O[2] = v8f{}; O[3] = v8f{};
  float m_i[8], l_i[8];
#pragma unroll
  for (int r = 0; r < 8; ++r) { m_i[r] = -1e30f; l_i[r] = 0.0f; }

  const int ldrow = tid >> 4;         // 0..15
  const int ldcol = (tid & 15) * 4;   // 0..60

  // Async-stage one 16x64 K tile + V tile (16 bytes per thread per tile).
  auto stage = [&](int kt_, int buf) {
    const size_t roff = (size_t)(kt_ * 16 + ldrow) * HEAD_DIM + ldcol;
    async_copy16(qbase + roff, &Kt[buf][ldrow * KSTRIDE + ldcol]);
    async_copy16(vbase + roff, &Vt[buf][ldrow * KSTRIDE + ldcol]);
  };

  constexpr int NT = SEQ / 16;
  stage(0, 0);  // prologue

  for (int kt = 0; kt < NT; ++kt) {
    const int cur = kt & 1;
    if (kt + 1 < NT) {
      stage(kt + 1, cur ^ 1);
      wait_async<2>();  // async loads complete in order: current tile is done
    } else {
      wait_async<0>();
    }
    __syncthreads();

    // S = Q @ K^T (keys are Q rows): 16 fp32 WMMAs.
    v8f c = v8f{};
#pragma unroll
    for (int kc = 0; kc < 16; ++kc) {
      v2f bf = *(const v2f*)(&Kt[cur][l16 * KSTRIDE + kc * 4 + half * 2]);
      c = wmma_f32(aq[kc], bf, c);
    }

    const int   ck  = cbase[kt * 16 + l16];
    const float amv = am[kt * 16 + l16];

    float p[8], alpha[8];
#pragma unroll
    for (int r = 0; r < 8; ++r) {
      float sv = (c[r] + ((cq[r] == ck) ? 0.0f : NEGBIG)) * 0.125f + amv;
      // row max across the 16 lanes of this half-group
      float mx = sv;
      mx = fmaxf(mx, __shfl_xor(mx, 1, 32));
      mx = fmaxf(mx, __shfl_xor(mx, 2, 32));
      mx = fmaxf(mx, __shfl_xor(mx, 4, 32));
      mx = fmaxf(mx, __shfl_xor(mx, 8, 32));
      const float mn = fmaxf(m_i[r], mx);
      const float al = __expf(m_i[r] - mn);
      const float pv = __expf(sv - mn);
      float rs = pv;
      rs += __shfl_xor(rs, 1, 32);
      rs += __shfl_xor(rs, 2, 32);
      rs += __shfl_xor(rs, 4, 32);
      rs += __shfl_xor(rs, 8, 32);
      l_i[r]   = l_i[r] * al + rs;
      m_i[r]   = mn;
      p[r]     = pv;
      alpha[r] = al;
    }
#pragma unroll
    for (int t = 0; t < 4; ++t)
#pragma unroll
      for (int r = 0; r < 8; ++r) O[t][r] *= alpha[r];

    // Re-layout P (C-layout -> A-layout) via this wave's private LDS slab.
    float* pl = Pl[wave];
#pragma unroll
    for (int r = 0; r < 8; ++r) pl[(half * 8 + r) * PSTRIDE + l16] = p[r];
    __asm__ volatile("" ::: "memory");  // wave is lockstep; LDS is in-order
    v2f ap[4];
    {
      const float* pr = pl + l16 * PSTRIDE + half * 2;
#pragma unroll
      for (int kc = 0; kc < 4; ++kc) ap[kc] = *(const v2f*)(pr + kc * 4);
    }

    // O += P @ V : 16 fp32 WMMAs.
#pragma unroll
    for (int kc = 0; kc < 4; ++kc) {
      const float* vr0 = &Vt[cur][(kc * 4 + half * 2) * KSTRIDE];
      const float* vr1 = vr0 + KSTRIDE;
#pragma unroll
      for (int t = 0; t < 4; ++t) {
        v2f bf;
        bf.x = vr0[t * 16 + l16];
        bf.y = vr1[t * 16 + l16];
        O[t] = wmma_f32(ap[kc], bf, O[t]);
      }
    }
    __syncthreads();  // all waves done with `cur` before it is re-staged
  }

  // Epilogue: normalize and write [B,S,NH*D].
#pragma unroll
  for (int t = 0; t < 4; ++t) {
    const int cc = t * 16 + l16;
#pragma unroll
    for (int r = 0; r < 8; ++r) {
      const int s = qrow0 + half * 8 + r;
      out[(size_t)(b * SEQ + s) * HID + h * HEAD_DIM + cc] = O[t][r] / l_i[r];
    }
  }
}

// ---------------------------------------------------------------------------
extern "C" void kernel_launch(void* const* d_in, const int* in_sizes, int n_in,
                              void* d_out, int out_size, void* d_ws,
                              size_t ws_size, hipStream_t stream) {
  const float* hs    = (const float*)d_in[0];
  const float* amask = (const float*)d_in[1];
  const float* Wq    = (const float*)d_in[2];
  const float* bq    = (const float*)d_in[3];
  const float* Wv    = (const float*)d_in[4];
  const float* bv    = (const float*)d_in[5];
  const float* lsh   = (const float*)d_in[6];

  const size_t qv_elems = (size_t)BATCH * NUM_HEADS * SEQ * HEAD_DIM;  // 3.1M
  float* Qh   = (float*)d_ws;
  float* Vh   = Qh + qv_elems;
  int* codes  = (int*)(Vh + qv_elems);

  qv_proj_kernel<<<dim3((BATCH * SEQ) / 16, HID / 64, 2), 32, 0, stream>>>(
      hs, Wq, bq, Wv, bv, Qh, Vh);
  lsh_codes_kernel<<<dim3(BATCH * NUM_HEADS), 256, 0, stream>>>(Qh, lsh, codes);
  attn_kernel<<<dim3(SEQ / 128, NUM_HEADS, BATCH), 256, 0, stream>>>(
      Qh, Vh, codes, amask, (float*)d_out);
}